// FCN_7164005450287
// MI455X (gfx1250) — compile-verified
//
#include <hip/hip_runtime.h>

typedef __attribute__((ext_vector_type(2))) float v2f;
typedef __attribute__((ext_vector_type(8))) float v8f;
typedef __attribute__((ext_vector_type(4))) unsigned int v4u;
typedef __attribute__((ext_vector_type(8))) int v8i;
typedef __attribute__((ext_vector_type(4))) int v4i;

#define NPTS_ 101

// ds_swizzle group-of-32: and=0x1f, or=0, xor=0x10  ->  lane ^ 16
__device__ __forceinline__ float swz16(float v) {
    return __int_as_float(__builtin_amdgcn_ds_swizzle(__float_as_int(v), 0x401F));
}

// TDM: DMA one contiguous row of `nelem` f32 from global into LDS at byte
// offset lds_off. Descriptor packed per CDNA5 ISA D# layout (CDNA5 ISA).
__device__ __forceinline__ void tdm_load_row_to_lds(const void* gaddr, unsigned lds_off, int nelem) {
    unsigned long long ga = (unsigned long long)gaddr;
    v4u g0;
    g0.x = 1u;                                                // count=1 (valid user D#)
    g0.y = lds_off;                                           // lds_addr (bytes)
    g0.z = (unsigned)(ga & 0xffffffffu);                      // global_addr[31:0]
    g0.w = (unsigned)((ga >> 32) & 0x01ffffffu) | 0x80000000u; // addr[56:32] | type=2
    v8i g1;
    unsigned n = (unsigned)nelem;
    g1[0] = 0x00020000;                       // wg_mask=0, data_size=2 (4 bytes)
    g1[1] = (int)((n & 0xffffu) << 16);       // tensor_dim0[15:0] at bits[63:48]
    g1[2] = (int)(((n >> 16) & 0xffffu) | (1u << 16)); // dim0[31:16], tensor_dim1=1
    g1[3] = (int)((n & 0xffffu) << 16);       // tensor_dim1 hi=0, tile_dim0=n
    g1[4] = 1;                                // tile_dim1=1, tile_dim2=0
    g1[5] = (int)n;                           // tensor_dim0_stride[31:0]
    g1[6] = 0;                                // stride0 hi, stride1 lo
    g1[7] = 0;
    v4i z4 = {0, 0, 0, 0};
#if __clang_major__ >= 23
    v8i z8 = {0, 0, 0, 0, 0, 0, 0, 0};
    __builtin_amdgcn_tensor_load_to_lds(g0, g1, z4, z4, z8, 0);
#else
    __builtin_amdgcn_tensor_load_to_lds(g0, g1, z4, z4, 0);
#endif
}

// Partial GEMV: y_part[ks, row] = sum_{k in chunk ks} W[row,k] * x[k].
// W row-major (R x K). One wave -> 16 rows; blockIdx.y -> K-chunk (size Kc).
// Uses V_WMMA_F32_16X16X4_F32: A = x broadcast to all 16 M-rows, B = W tile.
__global__ __launch_bounds__(128)
void gemv_wmma_kernel(const float* __restrict__ W, const float* __restrict__ xin,
                      float* __restrict__ part, int R, int K, int Kc)
{
    extern __shared__ float s_x[];
    const int tid  = threadIdx.x;
    const int Kbeg = blockIdx.y * Kc;

    // Stage x chunk into LDS with the Tensor Data Mover (wave 0 issues the DMA).
    if (tid < 32) {
        tdm_load_row_to_lds(xin + Kbeg, 0u, Kc);
        __builtin_amdgcn_s_wait_tensorcnt(0);
    }
    __syncthreads();
    // Correctness hedge (TDM descriptor semantics not HW-verified): rewrite the
    // same values manually. Negligible traffic vs the W stream.
    for (int i = tid; i < Kc; i += 128) s_x[i] = xin[Kbeg + i];
    __syncthreads();

    const int lane = tid & 31;
    const int wave = tid >> 5;
    const int rowTile = (blockIdx.x * 4 + wave) * 16;
    if (rowTile >= R) return;                 // wave-uniform exit (partial last block)

    int r = rowTile + (lane & 15);
    if (r > R - 1) r = R - 1;                 // clamp: keep EXEC all-1s for WMMA
    const int half = lane >> 4;               // 0: lanes 0-15, 1: lanes 16-31
    const float* Wrow = W + (size_t)r * (size_t)K + (size_t)(Kbeg + 4 * half);

    v8f acc = {0.f, 0.f, 0.f, 0.f, 0.f, 0.f, 0.f, 0.f};

    #pragma unroll 4
    for (int kk = 0; kk < Kc; kk += 8) {
        float4 w4 = *(const float4*)(Wrow + kk);          // 16 rows x 8 K per wave
        __builtin_prefetch((const void*)(Wrow + kk + 256), 0, 1); // global_prefetch_b8

        // A fragments: 16x4 layout -> lanes0-15:(K0,K1), lanes16-31:(K2,K3)
        const int ax = kk + (half << 1);
        v2f a1 = { s_x[ax],     s_x[ax + 1] };            // K = kk..kk+3
        v2f a2 = { s_x[ax + 4], s_x[ax + 5] };            // K = kk+4..kk+7

        // exchange halves (lane ^ 16) to build B 4x16 fragments
        float sx = swz16(w4.x), sy = swz16(w4.y);
        float sz = swz16(w4.z), sw = swz16(w4.w);
        const bool lo = (half == 0);
        v2f b1 = { lo ? w4.x : sz, lo ? w4.y : sw };      // B for K kk..kk+3
        v2f b2 = { lo ? sx : w4.z, lo ? sy : w4.w };      // B for K kk+4..kk+7

        acc = __builtin_amdgcn_wmma_f32_16x16x4_f32(false, a1, false, b1,
                                                    (short)0, acc, false, false);
        acc = __builtin_amdgcn_wmma_f32_16x16x4_f32(false, a2, false, b2,
                                                    (short)0, acc, false, false);
    }

    // D row M=0 (VGPR0, lanes 0-15) holds the partial GEMV results for this tile
    if (lane < 16) {
        const int row = rowTile + lane;
        if (row < R) part[(size_t)blockIdx.y * (size_t)R + row] = acc[0];
    }
}

// Deterministic K-split reduction + bias + optional relu (no float atomics).
__global__ __launch_bounds__(256)
void reduce_bias_act_kernel(const float* __restrict__ part, const float* __restrict__ bias,
                            float* __restrict__ yout, int R, int nsplit, int relu)
{
    const int r = blockIdx.x * 256 + threadIdx.x;
    if (r < R) {
        float s = bias[r];
        for (int k = 0; k < nsplit; ++k) s += part[(size_t)k * (size_t)R + r];
        if (relu) s = fmaxf(s, 0.0f);
        yout[r] = s;
    }
}

__global__ __launch_bounds__(256)
void sigma_kernel(const float* __restrict__ fc, float* __restrict__ Sigma)
{
    const int i = blockIdx.x * 256 + threadIdx.x;
    if (i < 4097) {
        float s = 0.f;
        #pragma unroll
        for (int k = 0; k < 10; ++k) s += fc[i + k];
        Sigma[i] = s * 0.1f;
    }
}

__global__ __launch_bounds__(256)
void last_kernel(const float* __restrict__ Sigma, float* __restrict__ out_last)
{
    __shared__ float red[256];
    const int tid = threadIdx.x;
    float s = 0.f;
    for (int i = tid; i < 4097; i += 256) s += fminf(Sigma[i], 0.f);
    red[tid] = s;
    __syncthreads();
    for (int off = 128; off > 0; off >>= 1) {
        if (tid < off) red[tid] += red[tid + off];
        __syncthreads();
    }
    if (tid == 0) out_last[0] = red[0];
}

// One block = one s (256 i's); Sigma + integration tables staged in LDS.
__global__ __launch_bounds__(256)
void post_kernel(const float* __restrict__ xin, const float* __restrict__ fc,
                 const float* __restrict__ Sigma, const int* __restrict__ pWc,
                 float* __restrict__ out)
{
    __shared__ float sSig[4097];
    __shared__ float sgw[NPTS_];
    __shared__ float sgtw[NPTS_];

    const float Wc = (float)(*pWc);
    const int tid = threadIdx.x;

    // TDM-stage Sigma into LDS (sSig is the first/offset-0 LDS object), then hedge.
    if (tid < 32) {
        tdm_load_row_to_lds(Sigma, 0u, 4097);
        __builtin_amdgcn_s_wait_tensorcnt(0);
    }
    __syncthreads();
    for (int i = tid; i < 4097; i += 256) sSig[i] = Sigma[i];
    if (tid < NPTS_) {
        const float t  = -1.0f + (float)tid * (2.0f / (float)(NPTS_ - 1));
        const float eu = expf(Wc * t);            // u = newE/x = Wc*t (x-independent)
        const float g  = eu / ((eu + 1.0f) * (eu + 1.0f));
        const float wt = (tid == 0 || tid == NPTS_ - 1) ? 0.5f : 1.0f;
        sgw[tid]  = g * wt;                        // trapz weight folded in
        sgtw[tid] = g * t * wt;
    }
    __syncthreads();

    const int s = blockIdx.x >> 2;
    const int i = ((blockIdx.x & 3) << 8) + tid;   // 0..1023
    const float xi = xin[i];
    const float A  = Wc * xi;                      // newE = A * t[j]
    const float c0 = fc[4106 + 3 * s];
    const float c2 = fc[4106 + 3 * s + 2];
    const float chem = atanf(fmaf(c2, xi * xi, c0)) * (Wc * 0.63661977236758134f); // *2/pi
    const float invde = 1024.0f / Wc;              // 4096 / (4*Wc), uniform grid
    const float e0 = -2.0f * Wc;

    float acc1 = 0.f, acc2 = 0.f;
    const float dt = 2.0f / (float)(NPTS_ - 1);
    for (int j = 0; j < NPTS_; ++j) {
        const float t = -1.0f + (float)j * dt;
        const float q = fmaf(A, t, chem);
        float idxf = (q - e0) * invde;
        idxf = fminf(fmaxf(idxf, 0.0f), 4096.0f);
        int i0 = (int)idxf;
        if (i0 > 4095) i0 = 4095;
        const float fr = idxf - (float)i0;
        const float s0 = sSig[i0];
        const float S  = fmaf(sSig[i0 + 1] - s0, fr, s0);  // jnp.interp (clamped)
        acc1 = fmaf(S, sgw[j],  acc1);
        acc2 = fmaf(S, sgtw[j], acc2);
    }
    const float C = 2.0f * Wc / (float)(NPTS_ - 1);   // dx * (1/x) cancels x
    out[(size_t)s * 2048 + i]        =  C * acc1;            // I1
    out[(size_t)s * 2048 + 1024 + i] = -(C * Wc * xi * acc2); // -I2
}

extern "C" void kernel_launch(void* const* d_in, const int* in_sizes, int n_in,
                              void* d_out, int out_size, void* d_ws, size_t ws_size,
                              hipStream_t stream)
{
    (void)in_sizes; (void)n_in; (void)out_size; (void)ws_size;
    const float* x  = (const float*)d_in[0];
    const float* W0 = (const float*)d_in[1];
    const float* b0 = (const float*)d_in[2];
    const float* W1 = (const float*)d_in[3];
    const float* b1 = (const float*)d_in[4];
    const float* W2 = (const float*)d_in[5];
    const float* b2 = (const float*)d_in[6];
    const int*  pWc = (const int*)d_in[7];
    float* out = (float*)d_out;
    float* ws  = (float*)d_ws;

    float* h0   = ws;             // 4096
    float* h1   = ws + 4096;      // 4096
    float* fc   = ws + 8192;      // 4874
    float* Sg   = ws + 13312;     // 4097
    float* part = ws + 17664;     // up to 4 * 4874 partials

    // layer 0: 4096 x 1024, K-split x2
    gemv_wmma_kernel<<<dim3(64, 2), 128, 512 * sizeof(float), stream>>>(W0, x, part, 4096, 1024, 512);
    reduce_bias_act_kernel<<<16, 256, 0, stream>>>(part, b0, h0, 4096, 2, 1);
    // layer 1: 4096 x 4096, K-split x4
    gemv_wmma_kernel<<<dim3(64, 4), 128, 1024 * sizeof(float), stream>>>(W1, h0, part, 4096, 4096, 1024);
    reduce_bias_act_kernel<<<16, 256, 0, stream>>>(part, b1, h1, 4096, 4, 1);
    // layer 2: 4874 x 4096, K-split x4
    gemv_wmma_kernel<<<dim3(77, 4), 128, 1024 * sizeof(float), stream>>>(W2, h1, part, 4874, 4096, 1024);
    reduce_bias_act_kernel<<<20, 256, 0, stream>>>(part, b2, fc, 4874, 4, 0);

    sigma_kernel<<<17, 256, 0, stream>>>(fc, Sg);
    last_kernel<<<1, 256, 0, stream>>>(Sg, out + 524288);
    post_kernel<<<1024, 256, 0, stream>>>(x, fc, Sg, pWc, out);
}